// Quantizer_22162031247880
// MI455X (gfx1250) — compile-verified
//
#include <hip/hip_runtime.h>
#include <stdint.h>

typedef float v4f __attribute__((ext_vector_type(4)));

#define WAVES_PER_BLOCK 8
#define BLOCK_THREADS   (WAVES_PER_BLOCK * 32)   // 256 threads = 8 wave32
#define TILE            128                       // elems per wave-tile: 32 lanes x 4 floats
#define MAX_BLOCKS      4096

// ---------------------------------------------------------------------------
// CDNA5 async global->LDS copies (ASYNCcnt-tracked), 16 bytes per lane.
// First operand (VDST field) = per-lane LDS byte address, second = 64-bit
// global address, no SADDR. Low 32 bits of a flat LDS pointer are the LDS
// byte offset (ISA 10.2 aperture rules), so truncation is the LDS address.
//   _lu: last-use hint  -> x is L2-resident from pass 1, read once, discard.
//   _nt: non-temporal   -> noise streams from HBM exactly once, never allocate.
// ---------------------------------------------------------------------------
__device__ __forceinline__ void async_copy_b128_lu(const float* gsrc, float* ldst) {
  uint64_t ga = (uint64_t)(uintptr_t)gsrc;
  uint32_t la = (uint32_t)(uintptr_t)ldst;
  asm volatile("global_load_async_to_lds_b128 %0, %1, off th:TH_LOAD_LU"
               :: "v"(la), "v"(ga) : "memory");
}
__device__ __forceinline__ void async_copy_b128_nt(const float* gsrc, float* ldst) {
  uint64_t ga = (uint64_t)(uintptr_t)gsrc;
  uint32_t la = (uint32_t)(uintptr_t)ldst;
  asm volatile("global_load_async_to_lds_b128 %0, %1, off th:TH_LOAD_NT"
               :: "v"(la), "v"(ga) : "memory");
}
__device__ __forceinline__ void wait_async_le2() {
  asm volatile("s_wait_asynccnt 2" ::: "memory");
}
__device__ __forceinline__ void wait_async_le0() {
  asm volatile("s_wait_asynccnt 0" ::: "memory");
}

// ---------------------------------------------------------------------------
// E5M2 stochastic-rounding quantize of one element (MAN=2, EXP=5).
//   e       = clamp(floor(log2|xs|), -14, 15)   -> float exponent field
//   step    = 2^(e-2), 1/step = 2^(2-e)         -> exact bit-built powers of 2
//   q       = min(floor(|xs|/step + noise) * step, 57344)
//   result  = sign(xs) * q / factor             (factor is a power of 2: exact)
// ---------------------------------------------------------------------------
__device__ __forceinline__ float quantize_one(float xv, float nz, float factor, float invfac) {
  float xs = xv * factor;
  float ax = fabsf(xs);
  int e = (int)(__float_as_uint(ax) >> 23) - 127;   // floor(log2) for normals; subnormals clamp
  e = e < -14 ? -14 : (e > 15 ? 15 : e);
  float stepinv = __uint_as_float((uint32_t)(129 - e) << 23);  // 2^(2-e), exp field in [114,143]
  float step    = __uint_as_float((uint32_t)(125 + e) << 23);  // 2^(e-2), exp field in [111,140]
  float q = floorf(ax * stepinv + nz) * step;                  // ax*stepinv is exact
  q = fminf(q, 57344.0f);                                      // (2 - 2^-2) * 2^15
  return copysignf(q, xs) * invfac;
}

// ---------------------------------------------------------------------------
// Pass 0: zero the absmax accumulator in workspace (d_ws is poisoned).
// ---------------------------------------------------------------------------
__global__ void ws_init(unsigned* ws) { ws[0] = 0u; }

// ---------------------------------------------------------------------------
// Pass 1: grid-wide absmax. b128 loads with default RT hint: deliberately warm
// the 192MB L2 with x (84MB) so pass 2 re-reads it from L2, not HBM.
// Wave32 shuffle reduce -> LDS -> one uint atomicMax per block.
// ---------------------------------------------------------------------------
__global__ void __launch_bounds__(BLOCK_THREADS)
absmax_reduce(const float* __restrict__ x, unsigned* __restrict__ ws, int n) {
  float m = 0.0f;
  const int gtid    = blockIdx.x * BLOCK_THREADS + threadIdx.x;
  const int gstride = gridDim.x * BLOCK_THREADS;
  const int n4 = n >> 2;
  const v4f* x4 = (const v4f*)x;
  for (int i = gtid; i < n4; i += gstride) {
    v4f v = x4[i];
    m = fmaxf(m, fmaxf(fmaxf(fabsf(v.x), fabsf(v.y)), fmaxf(fabsf(v.z), fabsf(v.w))));
  }
  for (int i = (n4 << 2) + gtid; i < n; i += gstride) m = fmaxf(m, fabsf(x[i]));
#pragma unroll
  for (int off = 16; off > 0; off >>= 1) m = fmaxf(m, __shfl_xor(m, off, 32));
  __shared__ float sm[WAVES_PER_BLOCK];
  if ((threadIdx.x & 31) == 0) sm[threadIdx.x >> 5] = m;
  __syncthreads();
  if (threadIdx.x == 0) {
    float bm = sm[0];
#pragma unroll
    for (int w = 1; w < WAVES_PER_BLOCK; ++w) bm = fmaxf(bm, sm[w]);
    atomicMax(ws, __float_as_uint(bm));   // valid: values are non-negative floats
  }
}

// ---------------------------------------------------------------------------
// Pass 2: per-wave double-buffered async pipeline.
//   stage = 2 async b128 copies (x tile LU-hinted, noise tile NT-hinted,
//   512B each per wave). Issue next stage -> s_wait_asynccnt 2 (async loads
//   complete in order, so <=2 of 4 outstanding means the current buffer
//   landed) -> ds_load_b128, quantize, nontemporal b128 store to out.
// ---------------------------------------------------------------------------
__global__ void __launch_bounds__(BLOCK_THREADS)
fp8e5m2_sr_quant(const float* __restrict__ x, const float* __restrict__ noise,
                 float* __restrict__ out, const unsigned* __restrict__ absmax_bits,
                 int n) {
  __shared__ float lx[WAVES_PER_BLOCK][2][TILE];
  __shared__ float ln[WAVES_PER_BLOCK][2][TILE];

  const float absmax = __uint_as_float(*absmax_bits);
  const int   k      = (int)floorf(16.0f - log2f(absmax * (1.0f / 1.75f)));
  const float factor = ldexpf(1.0f, k);
  const float invfac = ldexpf(1.0f, -k);

  const int lane       = threadIdx.x & 31;
  const int wave       = threadIdx.x >> 5;
  const int nTiles     = n >> 7;
  const int waveStride = gridDim.x * WAVES_PER_BLOCK;
  int t = blockIdx.x * WAVES_PER_BLOCK + wave;

  if (t < nTiles) {
    const int    base = lane << 2;                  // float index of this lane's 16B chunk
    const float* gx   = x     + base;               // per-lane base pointers; advance by
    const float* gn   = noise + base;               // tile*TILE floats per stage
    float*       go   = out   + base;
    float* lx0 = &lx[wave][0][base];
    float* lx1 = &lx[wave][1][base];
    float* ln0 = &ln[wave][0][base];
    float* ln1 = &ln[wave][1][base];

    int buf = 0;
    async_copy_b128_lu(gx + (size_t)t * TILE, lx0);
    async_copy_b128_nt(gn + (size_t)t * TILE, ln0);
    for (;;) {
      const int  tn      = t + waveStride;
      const bool hasNext = tn < nTiles;
      if (hasNext) {
        async_copy_b128_lu(gx + (size_t)tn * TILE, buf ? lx0 : lx1);
        async_copy_b128_nt(gn + (size_t)tn * TILE, buf ? ln0 : ln1);
        wait_async_le2();
      } else {
        wait_async_le0();
      }
      v4f xv = *(const v4f*)(buf ? lx1 : lx0);      // ds_load_b128
      v4f nv = *(const v4f*)(buf ? ln1 : ln0);
      v4f r;
      r.x = quantize_one(xv.x, nv.x, factor, invfac);
      r.y = quantize_one(xv.y, nv.y, factor, invfac);
      r.z = quantize_one(xv.z, nv.z, factor, invfac);
      r.w = quantize_one(xv.w, nv.w, factor, invfac);
      __builtin_nontemporal_store(r, (v4f*)(go + (size_t)t * TILE));
      if (!hasNext) break;
      t = tn;
      buf ^= 1;
    }
  }

  // Tail (n % 128 != 0): dead for the given shape (20,971,520 % 128 == 0),
  // kept for safety with simple guarded scalar loads.
  const int gtid    = blockIdx.x * BLOCK_THREADS + threadIdx.x;
  const int gstride = gridDim.x * BLOCK_THREADS;
  for (int i = (nTiles << 7) + gtid; i < n; i += gstride)
    out[i] = quantize_one(x[i], noise[i], factor, invfac);
}

// ---------------------------------------------------------------------------
extern "C" void kernel_launch(void* const* d_in, const int* in_sizes, int n_in,
                              void* d_out, int out_size, void* d_ws, size_t ws_size,
                              hipStream_t stream) {
  const float* x     = (const float*)d_in[0];
  const float* noise = (const float*)d_in[1];
  float*       out   = (float*)d_out;
  unsigned*    ws    = (unsigned*)d_ws;
  const int    n     = in_sizes[0];

  ws_init<<<1, 1, 0, stream>>>(ws);

  int b1 = (n / 4 + BLOCK_THREADS - 1) / BLOCK_THREADS;
  if (b1 < 1) b1 = 1;
  if (b1 > MAX_BLOCKS) b1 = MAX_BLOCKS;
  absmax_reduce<<<b1, BLOCK_THREADS, 0, stream>>>(x, ws, n);

  const int nTiles = n >> 7;
  int b2 = (nTiles + WAVES_PER_BLOCK - 1) / WAVES_PER_BLOCK;
  if (b2 < 1) b2 = 1;
  if (b2 > MAX_BLOCKS) b2 = MAX_BLOCKS;
  fp8e5m2_sr_quant<<<b2, BLOCK_THREADS, 0, stream>>>(x, noise, out, ws, n);
}